// RippleNet_38611755991250
// MI455X (gfx1250) — compile-verified
//
#include <hip/hip_runtime.h>
#include <cstddef>

#define BATCH 4096
#define NMEM  256
#define DIM   64
#define ROWS  16          // batch rows per workgroup (one WMMA M-tile)
#define TPB   256         // 8 wave32

typedef float v2f __attribute__((ext_vector_type(2)));
typedef float v8f __attribute__((ext_vector_type(8)));

// One hop: item' = (item + softmax((h*r)·item) @ t) @ W^T
// Block: 16 batch rows. 8 waves: waves own 2 rows each for attention,
// waves 0..3 each own one 16-col WMMA tile of the 16x64 GEMM.
__global__ __launch_bounds__(TPB) void ripple_hop_kernel(
    const int*   __restrict__ item_ids,
    const float* __restrict__ item_in,
    const int*   __restrict__ h_idx,
    const int*   __restrict__ r_idx,
    const int*   __restrict__ t_idx,
    const float* __restrict__ ent,
    const float* __restrict__ rel,
    const float* __restrict__ W,
    float*       __restrict__ out,
    int use_ids)
{
    __shared__ float x_s[ROWS][DIM];        // item vectors (broadcast reads, no conflict)
    __shared__ float y_s[ROWS][DIM + 1];    // item + o, padded: conflict-free column reads for WMMA A

    const int tid  = threadIdx.x;
    const int wave = tid >> 5;
    const int lane = tid & 31;
    const int row0 = blockIdx.x * ROWS;

    // ---- Phase 1: stage 16 item vectors into LDS (256 thr x 4 floats = 16x64) ----
    {
        const int idx = tid * 4;
        const int r   = idx >> 6;
        const int c   = idx & 63;
        const float* src = use_ids ? (ent + (size_t)item_ids[row0 + r] * DIM)
                                   : (item_in + (size_t)(row0 + r) * DIM);
        float4 v = *(const float4*)(src + c);
        x_s[r][c]     = v.x;
        x_s[r][c + 1] = v.y;
        x_s[r][c + 2] = v.z;
        x_s[r][c + 3] = v.w;
    }
    __syncthreads();

    // ---- Phase 2: attention + weighted sum; each wave handles 2 rows ----
    for (int rr = 0; rr < 2; ++rr) {
        const int row = wave * 2 + rr;
        const int b   = row0 + row;
        const int* hI = h_idx + (size_t)b * NMEM;
        const int* rI = r_idx + (size_t)b * NMEM;
        const int* tI = t_idx + (size_t)b * NMEM;

        int   tRow[8];
        float logit[8];
        #pragma unroll
        for (int mi = 0; mi < 8; ++mi) {
            const int m = mi * 32 + lane;              // lane owns 8 memories
            tRow[mi] = tI[m];
            __builtin_prefetch(ent + (size_t)tRow[mi] * DIM, 0, 0);  // global_prefetch_b8
            const float4* hp = (const float4*)(ent + (size_t)hI[m] * DIM);
            const float4* rp = (const float4*)(rel + (size_t)rI[m] * DIM);
            const float4* xp = (const float4*)(&x_s[row][0]);
            float acc = 0.f;
            #pragma unroll
            for (int d4 = 0; d4 < DIM / 4; ++d4) {
                const float4 hv = hp[d4];
                const float4 rv = rp[d4];
                const float4 xv = xp[d4];
                acc = fmaf(hv.x * rv.x, xv.x, acc);
                acc = fmaf(hv.y * rv.y, xv.y, acc);
                acc = fmaf(hv.z * rv.z, xv.z, acc);
                acc = fmaf(hv.w * rv.w, xv.w, acc);
            }
            logit[mi] = acc;
        }

        // softmax over the 256 logits owned by this wave (8 per lane)
        float mx = logit[0];
        #pragma unroll
        for (int mi = 1; mi < 8; ++mi) mx = fmaxf(mx, logit[mi]);
        #pragma unroll
        for (int off = 16; off > 0; off >>= 1) mx = fmaxf(mx, __shfl_xor(mx, off, 32));

        float p[8];
        float ssum = 0.f;
        #pragma unroll
        for (int mi = 0; mi < 8; ++mi) { p[mi] = __expf(logit[mi] - mx); ssum += p[mi]; }
        #pragma unroll
        for (int off = 16; off > 0; off >>= 1) ssum += __shfl_xor(ssum, off, 32);
        const float inv = 1.0f / ssum;

        // o[d] = sum_m p[m] * t[m][d]; broadcast (p, t-index) via shfl so every
        // t row is read fully coalesced (lanes cover d = 0..63 in two 128B lines)
        float o0 = 0.f, o1 = 0.f;
        #pragma unroll
        for (int mi = 0; mi < 8; ++mi) {
            const float pv_src = p[mi];
            const int   tr_src = tRow[mi];
            for (int j = 0; j < 32; ++j) {
                const float pv = __shfl(pv_src, j, 32);
                const int   tr = __shfl(tr_src, j, 32);
                const float* tp = ent + (size_t)tr * DIM;
                o0 = fmaf(pv, tp[lane],      o0);
                o1 = fmaf(pv, tp[lane + 32], o1);
            }
        }
        y_s[row][lane]      = x_s[row][lane]      + o0 * inv;
        y_s[row][lane + 32] = x_s[row][lane + 32] + o1 * inv;
    }
    __syncthreads();

    // ---- Phase 3: out[16x64] = y[16x64] @ W^T via fp32 WMMA 16x16x4 ----
    // out[m][j] = sum_d y[m][d] * W[j][d]  ->  A[M][K]=y, B[K][N]=W[N][K]
#if __has_builtin(__builtin_amdgcn_wmma_f32_16x16x4_f32)
    if (wave < 4) {                       // wave-uniform branch: EXEC all-1s inside
        const int nt   = wave;            // N-tile (cols nt*16 .. nt*16+15)
        const int half = lane >> 4;       // lanes 16..31 hold K+2,K+3 / M+8 rows
        const int l16  = lane & 15;
        const int col  = nt * 16 + l16;
        v8f c = {0.f, 0.f, 0.f, 0.f, 0.f, 0.f, 0.f, 0.f};
        #pragma unroll
        for (int k0 = 0; k0 < DIM; k0 += 4) {
            const int kk = k0 + half * 2;
            v2f a, bm;
            a.x  = y_s[l16][kk];                     // A: M=l16, K=kk
            a.y  = y_s[l16][kk + 1];                 // A: M=l16, K=kk+1
            bm.x = W[(size_t)col * DIM + kk];        // B: K=kk,   N=col
            bm.y = W[(size_t)col * DIM + kk + 1];    // B: K=kk+1, N=col
            c = __builtin_amdgcn_wmma_f32_16x16x4_f32(
                    false, a, false, bm, (short)0, c, false, false);
        }
        #pragma unroll
        for (int v = 0; v < 8; ++v) {
            const int mrow = v + half * 8;           // C/D layout: VGPR v -> M=v(+8)
            out[(size_t)(row0 + mrow) * DIM + col] = c[v];
        }
    }
#else
    // scalar fallback (should not be taken on gfx1250)
    if (wave < 4) {
        const int nt   = wave;
        const int half = lane >> 4;
        const int l16  = lane & 15;
        const int col  = nt * 16 + l16;
        #pragma unroll
        for (int v = 0; v < 8; ++v) {
            const int mrow = v + half * 8;
            float acc = 0.f;
            for (int d = 0; d < DIM; ++d)
                acc = fmaf(y_s[mrow][d], W[(size_t)col * DIM + d], acc);
            out[(size_t)(row0 + mrow) * DIM + col] = acc;
        }
    }
#endif
}

// Final reduction: out[b] = sum_d item[b][d]; one wave per row.
__global__ __launch_bounds__(256) void ripple_rowsum_kernel(
    const float* __restrict__ in, float* __restrict__ outv)
{
    const int wave = threadIdx.x >> 5;
    const int lane = threadIdx.x & 31;
    const int b    = blockIdx.x * 8 + wave;
    const float* rp = in + (size_t)b * DIM;
    float s = rp[lane] + rp[lane + 32];
    #pragma unroll
    for (int off = 16; off > 0; off >>= 1) s += __shfl_xor(s, off, 32);
    if (lane == 0) outv[b] = s;
}

extern "C" void kernel_launch(void* const* d_in, const int* in_sizes, int n_in,
                              void* d_out, int out_size, void* d_ws, size_t ws_size,
                              hipStream_t stream) {
    const int*   item_ids = (const int*)  d_in[0];
    const int*   h0       = (const int*)  d_in[1];
    const int*   r0       = (const int*)  d_in[2];
    const int*   t0       = (const int*)  d_in[3];
    const int*   h1       = (const int*)  d_in[4];
    const int*   r1       = (const int*)  d_in[5];
    const int*   t1       = (const int*)  d_in[6];
    const float* ent      = (const float*)d_in[7];
    const float* rel      = (const float*)d_in[8];
    const float* W0       = (const float*)d_in[9];
    const float* W1       = (const float*)d_in[10];

    float* buf0 = (float*)d_ws;                       // [B, D] after hop 0
    float* buf1 = buf0 + (size_t)BATCH * DIM;         // [B, D] after hop 1

    dim3 grid(BATCH / ROWS);
    ripple_hop_kernel<<<grid, TPB, 0, stream>>>(
        item_ids, buf1 /*unused when use_ids=1*/, h0, r0, t0, ent, rel, W0, buf0, 1);
    ripple_hop_kernel<<<grid, TPB, 0, stream>>>(
        item_ids, buf0, h1, r1, t1, ent, rel, W1, buf1, 0);
    ripple_rowsum_kernel<<<BATCH / 8, 256, 0, stream>>>(buf1, (float*)d_out);
}